// ICP_56023553409627
// MI455X (gfx1250) — compile-verified
//
#include <hip/hip_runtime.h>
#include <hip/hip_bf16.h>
#include <math.h>

#define N_PTS 2048
#define BATCH 4
#define MAX_IT 10
#define TOL 1e-4f

typedef float v2f __attribute__((ext_vector_type(2)));
typedef float v8f __attribute__((ext_vector_type(8)));

// ---------------------------------------------------------------------------
// init: temp_pc = p1, prev_err = 0, done = 0
// ---------------------------------------------------------------------------
__global__ __launch_bounds__(256) void icp_init_kernel(const float* __restrict__ p1,
                                                       float* __restrict__ tpc,
                                                       float* __restrict__ prevErr,
                                                       int* __restrict__ doneFlag) {
    int i = blockIdx.x * 256 + threadIdx.x;
    if (i < BATCH * N_PTS * 3) tpc[i] = p1[i];
    if (i == 0) { prevErr[0] = 0.0f; doneFlag[0] = 0; }
}

// ---------------------------------------------------------------------------
// Nearest neighbor via V_WMMA_F32_16X16X4_F32.
// For each p2[b,i], find min_j / argmin_j || p2[b,i] - tpc[b,j] ||.
// A (16x4, M=j): row = (-2qx, -2qy, -2qz, |q|^2)
// B (4x16, N=i): col = ( px,  py,  pz,  1  )
// D[m][n] = |q_m|^2 - 2 p_n . q_m   (adding |p_n|^2 later; argmin unaffected)
//
// Only batch B-1's indices are ever consumed (reference uses idx[-1] for the
// gather), so batches 0..B-2 take a cheap min-only path (fmin trees), and only
// the last batch pays for compare/select argmin tracking.
//
// One wave handles 16 query points; workgroup (8 waves) = 128 queries of one
// batch; the whole batch's tpc is staged in LDS as float4.
// ---------------------------------------------------------------------------
__global__ __launch_bounds__(256) void icp_nn_kernel(const float* __restrict__ p2,
                                                     const float* __restrict__ tpc,
                                                     float* __restrict__ vals,
                                                     int* __restrict__ idx,
                                                     const int* __restrict__ doneFlag) {
    if (doneFlag[0]) return;

    __shared__ float4 sQ[N_PTS];  // 32 KB of the 320 KB WGP LDS pool

    const int groupsPerBatch = N_PTS / 128;
    const int b     = blockIdx.x / groupsPerBatch;
    const int chunk = blockIdx.x % groupsPerBatch;
    const float* q = tpc + (size_t)b * N_PTS * 3;
    const float* p = p2  + (size_t)b * N_PTS * 3;

    // stage temp_pc[b] with folded -2 scale and norm
    for (int j = threadIdx.x; j < N_PTS; j += 256) {
        float x = q[3 * j + 0], y = q[3 * j + 1], z = q[3 * j + 2];
        sQ[j] = make_float4(-2.0f * x, -2.0f * y, -2.0f * z, x * x + y * y + z * z);
    }
    __syncthreads();

    const int wave = threadIdx.x >> 5;      // 0..7
    const int lane = threadIdx.x & 31;      // wave32
    const int hi   = lane >> 4;             // lane half
    const int lm   = lane & 15;
    const int myI  = chunk * 128 + wave * 16 + lm;   // this lane's query point

    // B operand: one query point per column, loaded once
    float bx = p[3 * myI + 0], by = p[3 * myI + 1], bz = p[3 * myI + 2];
    float n2 = bx * bx + by * by + bz * bz;
    v2f Bv;
    if (hi == 0) { Bv.x = bx; Bv.y = by; }   // K=0,1
    else         { Bv.x = bz; Bv.y = 1.0f; } // K=2,3

    float bestD = 3.4e38f;

    if (b == BATCH - 1) {
        // ---- argmin path (indices consumed by the gather) ----
        int bestJ = 0;
        for (int jb = 0; jb < N_PTS; jb += 16) {
            float4 qa = sQ[jb + lm];                     // A row M=lm
            v2f Av;
            if (hi == 0) { Av.x = qa.x; Av.y = qa.y; }   // K=0,1
            else         { Av.x = qa.z; Av.y = qa.w; }   // K=2,3

            v8f C = {0.f, 0.f, 0.f, 0.f, 0.f, 0.f, 0.f, 0.f};
            C = __builtin_amdgcn_wmma_f32_16x16x4_f32(
                    false, Av, false, Bv, (short)0, C, false, false);

            const int jBase = jb + 8 * hi;               // rows M = v + 8*hi
#pragma unroll
            for (int v = 0; v < 8; ++v) {
                float d = C[v];
                int   j = jBase + v;
                if (d < bestD) { bestD = d; bestJ = j; }
            }
        }
        // merge the two lane halves (each column lives in lanes L and L+16)
        float oD = __shfl_xor(bestD, 16, 32);
        int   oJ = __shfl_xor(bestJ, 16, 32);
        if (oD < bestD || (oD == bestD && oJ < bestJ)) { bestD = oD; bestJ = oJ; }
        if (hi == 0) {
            vals[b * N_PTS + myI] = sqrtf(fmaxf(n2 + bestD, 0.0f));
            idx [b * N_PTS + myI] = bestJ;
        }
    } else {
        // ---- min-only path (vals feeds mean_err; idx never read) ----
        for (int jb = 0; jb < N_PTS; jb += 16) {
            float4 qa = sQ[jb + lm];
            v2f Av;
            if (hi == 0) { Av.x = qa.x; Av.y = qa.y; }
            else         { Av.x = qa.z; Av.y = qa.w; }

            v8f C = {0.f, 0.f, 0.f, 0.f, 0.f, 0.f, 0.f, 0.f};
            C = __builtin_amdgcn_wmma_f32_16x16x4_f32(
                    false, Av, false, Bv, (short)0, C, false, false);

            float m01 = fminf(C[0], C[1]);
            float m23 = fminf(C[2], C[3]);
            float m45 = fminf(C[4], C[5]);
            float m67 = fminf(C[6], C[7]);
            bestD = fminf(bestD, fminf(fminf(m01, m23), fminf(m45, m67)));
        }
        bestD = fminf(bestD, __shfl_xor(bestD, 16, 32));
        if (hi == 0) {
            vals[b * N_PTS + myI] = sqrtf(fmaxf(n2 + bestD, 0.0f));
        }
    }
}

// ---------------------------------------------------------------------------
// matched[b,k,:] = p2[b, idx[B-1, k], :]   (reference uses LAST batch's idx)
// ---------------------------------------------------------------------------
__global__ __launch_bounds__(256) void icp_gather_kernel(const float* __restrict__ p2,
                                                         const int* __restrict__ idx,
                                                         float* __restrict__ matched,
                                                         const int* __restrict__ doneFlag) {
    if (doneFlag[0]) return;
    int t = blockIdx.x * 256 + threadIdx.x;
    if (t >= BATCH * N_PTS) return;
    int b = t / N_PTS;
    int k = t - b * N_PTS;
    int j = idx[(BATCH - 1) * N_PTS + k];
    const float* src = p2 + ((size_t)b * N_PTS + j) * 3;
    float* dst = matched + (size_t)t * 3;
    dst[0] = src[0]; dst[1] = src[1]; dst[2] = src[2];
}

// ---------------------------------------------------------------------------
// 3x3 helpers (single-thread scalar code)
// ---------------------------------------------------------------------------
__device__ inline float det3(const float M[3][3]) {
    return M[0][0] * (M[1][1] * M[2][2] - M[1][2] * M[2][1])
         - M[0][1] * (M[1][0] * M[2][2] - M[1][2] * M[2][0])
         + M[0][2] * (M[1][0] * M[2][1] - M[1][1] * M[2][0]);
}

// SVD of 3x3: H = U diag(S) V^T, S descending.  Jacobi on H^T H.
__device__ inline void svd3(const float H[3][3], float U[3][3], float S[3], float V[3][3]) {
    float A[3][3];
    for (int i = 0; i < 3; i++)
        for (int j = 0; j < 3; j++) {
            float s = 0.f;
            for (int k = 0; k < 3; k++) s += H[k][i] * H[k][j];
            A[i][j] = s;
        }
    for (int i = 0; i < 3; i++)
        for (int j = 0; j < 3; j++) V[i][j] = (i == j) ? 1.f : 0.f;

    for (int sweep = 0; sweep < 16; sweep++) {
        for (int p = 0; p < 2; p++)
            for (int q = p + 1; q < 3; q++) {
                float apq = A[p][q];
                if (fabsf(apq) <= 1e-12f * (fabsf(A[p][p]) + fabsf(A[q][q]) + 1e-30f)) continue;
                float tau = (A[q][q] - A[p][p]) / (2.f * apq);
                float tt  = (tau >= 0.f ? 1.f : -1.f) / (fabsf(tau) + sqrtf(1.f + tau * tau));
                float cc  = 1.f / sqrtf(1.f + tt * tt);
                float ss  = tt * cc;
                for (int k = 0; k < 3; k++) {
                    float akp = A[k][p], akq = A[k][q];
                    A[k][p] = cc * akp - ss * akq;
                    A[k][q] = ss * akp + cc * akq;
                }
                for (int k = 0; k < 3; k++) {
                    float apk = A[p][k], aqk = A[q][k];
                    A[p][k] = cc * apk - ss * aqk;
                    A[q][k] = ss * apk + cc * aqk;
                }
                for (int k = 0; k < 3; k++) {
                    float vkp = V[k][p], vkq = V[k][q];
                    V[k][p] = cc * vkp - ss * vkq;
                    V[k][q] = ss * vkp + cc * vkq;
                }
            }
    }
    float w[3] = {fmaxf(A[0][0], 0.f), fmaxf(A[1][1], 0.f), fmaxf(A[2][2], 0.f)};
    int ord[3] = {0, 1, 2};
    if (w[ord[0]] < w[ord[1]]) { int t = ord[0]; ord[0] = ord[1]; ord[1] = t; }
    if (w[ord[1]] < w[ord[2]]) { int t = ord[1]; ord[1] = ord[2]; ord[2] = t; }
    if (w[ord[0]] < w[ord[1]]) { int t = ord[0]; ord[0] = ord[1]; ord[1] = t; }
    float Vs[3][3];
    for (int c = 0; c < 3; c++) {
        S[c] = sqrtf(w[ord[c]]);
        for (int r = 0; r < 3; r++) Vs[r][c] = V[r][ord[c]];
    }
    for (int r = 0; r < 3; r++)
        for (int c = 0; c < 3; c++) V[r][c] = Vs[r][c];

    for (int c = 0; c < 3; c++) {
        float u0 = 0.f, u1 = 0.f, u2 = 0.f;
        for (int k = 0; k < 3; k++) {
            u0 += H[0][k] * V[k][c];
            u1 += H[1][k] * V[k][c];
            u2 += H[2][k] * V[k][c];
        }
        float nrm = sqrtf(u0 * u0 + u1 * u1 + u2 * u2);
        if (nrm > 1e-12f) { u0 /= nrm; u1 /= nrm; u2 /= nrm; }
        U[0][c] = u0; U[1][c] = u1; U[2][c] = u2;
    }
    // rank-deficient fallback: third left-singular vector from cross product
    if (S[2] <= 1e-6f * (S[0] + 1e-30f)) {
        float cx = U[1][0] * U[2][1] - U[2][0] * U[1][1];
        float cy = U[2][0] * U[0][1] - U[0][0] * U[2][1];
        float cz = U[0][0] * U[1][1] - U[1][0] * U[0][1];
        float nrm = sqrtf(cx * cx + cy * cy + cz * cz);
        if (nrm > 1e-12f) { U[0][2] = cx / nrm; U[1][2] = cy / nrm; U[2][2] = cz / nrm; }
    }
}

// ---------------------------------------------------------------------------
// _get_transform(P1, P2): one workgroup per batch.
// H[j][k] = sum_n (P2[n]-c2)[j] * (P1[n]-c1)[k];  R = (U Vmod^T)^T, Vmod[2][2]*=d
// t = c2 - c1 @ R
// ---------------------------------------------------------------------------
__global__ __launch_bounds__(256) void icp_transform_kernel(const float* __restrict__ P1,
                                                            const float* __restrict__ P2,
                                                            float* __restrict__ Rout,
                                                            float* __restrict__ tout,
                                                            const int* __restrict__ doneFlag,
                                                            int checkDone) {
    if (checkDone && doneFlag[0]) return;
    const int b = blockIdx.x;
    const float* a = P1 + (size_t)b * N_PTS * 3;  // q1 source
    const float* c = P2 + (size_t)b * N_PTS * 3;  // q2 source

    float acc[15];
#pragma unroll
    for (int k = 0; k < 15; k++) acc[k] = 0.f;

    for (int n = threadIdx.x; n < N_PTS; n += 256) {
        float ax = a[3 * n], ay = a[3 * n + 1], az = a[3 * n + 2];
        float cx = c[3 * n], cy = c[3 * n + 1], cz = c[3 * n + 2];
        acc[0] += ax;  acc[1] += ay;  acc[2] += az;
        acc[3] += cx;  acc[4] += cy;  acc[5] += cz;
        acc[6]  += cx * ax; acc[7]  += cx * ay; acc[8]  += cx * az;
        acc[9]  += cy * ax; acc[10] += cy * ay; acc[11] += cy * az;
        acc[12] += cz * ax; acc[13] += cz * ay; acc[14] += cz * az;
    }
    // wave32 reduce, then across 8 waves through LDS
#pragma unroll
    for (int k = 0; k < 15; k++)
        for (int o = 16; o > 0; o >>= 1) acc[k] += __shfl_xor(acc[k], o, 32);

    __shared__ float sred[8][15];
    int wave = threadIdx.x >> 5, lane = threadIdx.x & 31;
    if (lane == 0)
        for (int k = 0; k < 15; k++) sred[wave][k] = acc[k];
    __syncthreads();

    if (threadIdx.x == 0) {
        float tot[15];
        for (int k = 0; k < 15; k++) {
            float s = 0.f;
            for (int w = 0; w < 8; w++) s += sred[w][k];
            tot[k] = s;
        }
        const float invN = 1.0f / (float)N_PTS;
        float c1[3] = {tot[0] * invN, tot[1] * invN, tot[2] * invN};
        float c2[3] = {tot[3] * invN, tot[4] * invN, tot[5] * invN};
        float H[3][3];
        for (int j = 0; j < 3; j++)
            for (int k = 0; k < 3; k++)
                H[j][k] = tot[6 + 3 * j + k] - (float)N_PTS * c2[j] * c1[k];

        float U[3][3], V[3][3], S[3];
        svd3(H, U, S, V);
        float d = det3(V) * det3(U);     // det(V @ U^T)
        V[2][2] *= d;                    // reference scales only element (2,2)
        float R[3][3];                   // R[i][j] = sum_k V[i][k] * U[j][k]
        for (int i = 0; i < 3; i++)
            for (int j = 0; j < 3; j++) {
                float s = 0.f;
                for (int k = 0; k < 3; k++) s += V[i][k] * U[j][k];
                R[i][j] = s;
            }
        float t[3];
        for (int k = 0; k < 3; k++) {
            float s = 0.f;
            for (int j = 0; j < 3; j++) s += c1[j] * R[j][k];
            t[k] = c2[k] - s;
        }
        for (int i = 0; i < 3; i++)
            for (int j = 0; j < 3; j++) Rout[b * 9 + i * 3 + j] = R[i][j];
        for (int k = 0; k < 3; k++) tout[b * 3 + k] = t[k];
    }
}

// ---------------------------------------------------------------------------
// temp_pc = temp_pc @ R + t   (row-vector convention)
// ---------------------------------------------------------------------------
__global__ __launch_bounds__(256) void icp_apply_kernel(float* __restrict__ tpc,
                                                        const float* __restrict__ R,
                                                        const float* __restrict__ t,
                                                        const int* __restrict__ doneFlag) {
    if (doneFlag[0]) return;
    int i = blockIdx.x * 256 + threadIdx.x;
    if (i >= BATCH * N_PTS) return;
    int b = i / N_PTS;
    const float* Rb = R + b * 9;
    const float* tb = t + b * 3;
    float x = tpc[3 * i], y = tpc[3 * i + 1], z = tpc[3 * i + 2];
    float nx = x * Rb[0] + y * Rb[3] + z * Rb[6] + tb[0];
    float ny = x * Rb[1] + y * Rb[4] + z * Rb[7] + tb[1];
    float nz = x * Rb[2] + y * Rb[5] + z * Rb[8] + tb[2];
    tpc[3 * i] = nx; tpc[3 * i + 1] = ny; tpc[3 * i + 2] = nz;
}

// ---------------------------------------------------------------------------
// mean error + convergence update. Runs LAST in each iteration so that
// gather/transform/apply of the same iteration used the pre-update flag.
// ---------------------------------------------------------------------------
__global__ __launch_bounds__(256) void icp_err_kernel(const float* __restrict__ vals,
                                                      float* __restrict__ prevErr,
                                                      int* __restrict__ doneFlag) {
    if (doneFlag[0]) return;
    __shared__ float sred[256];
    float s = 0.f;
    for (int i = threadIdx.x; i < BATCH * N_PTS; i += 256) s += vals[i];
    sred[threadIdx.x] = s;
    __syncthreads();
    for (int off = 128; off > 0; off >>= 1) {
        if (threadIdx.x < off) sred[threadIdx.x] += sred[threadIdx.x + off];
        __syncthreads();
    }
    if (threadIdx.x == 0) {
        float mean = sred[0] / (float)(BATCH * N_PTS);
        float prev = prevErr[0];
        doneFlag[0] = (fabsf(prev - mean) < TOL) ? 1 : 0;
        prevErr[0] = mean;
    }
}

// ---------------------------------------------------------------------------
// T = concat([R, t^T], axis=2)  -> (B, 3, 4)
// ---------------------------------------------------------------------------
__global__ void icp_writeT_kernel(const float* __restrict__ R,
                                  const float* __restrict__ t,
                                  float* __restrict__ out) {
    int i = threadIdx.x;
    if (i >= BATCH * 12) return;
    int b = i / 12;
    int r = (i % 12) / 4;
    int c = i % 4;
    out[i] = (c < 3) ? R[b * 9 + r * 3 + c] : t[b * 3 + r];
}

// ---------------------------------------------------------------------------
extern "C" void kernel_launch(void* const* d_in, const int* in_sizes, int n_in,
                              void* d_out, int out_size, void* d_ws, size_t ws_size,
                              hipStream_t stream) {
    (void)in_sizes; (void)n_in; (void)out_size; (void)ws_size;
    const float* p1 = (const float*)d_in[0];
    const float* p2 = (const float*)d_in[1];
    float* out = (float*)d_out;

    char* ws = (char*)d_ws;
    float* tpc     = (float*)(ws + 0);          // 4*2048*3 f = 96 KB
    float* matched = (float*)(ws + 98304);      // 96 KB
    float* vals    = (float*)(ws + 196608);     // 32 KB
    int*   idx     = (int*)  (ws + 229376);     // 32 KB (only last batch written)
    float* Rw      = (float*)(ws + 262144);     // 36 f
    float* tw      = (float*)(ws + 262400);     // 12 f
    float* prevErr = (float*)(ws + 262528);
    int*   done    = (int*)  (ws + 262532);

    icp_init_kernel<<<(BATCH * N_PTS * 3 + 255) / 256, 256, 0, stream>>>(p1, tpc, prevErr, done);

    const int nnBlocks  = BATCH * (N_PTS / 128);        // 64
    const int ptBlocks  = (BATCH * N_PTS + 255) / 256;  // 32

    for (int it = 0; it <= MAX_IT; ++it) {
        icp_nn_kernel<<<nnBlocks, 256, 0, stream>>>(p2, tpc, vals, idx, done);
        icp_gather_kernel<<<ptBlocks, 256, 0, stream>>>(p2, idx, matched, done);
        icp_transform_kernel<<<BATCH, 256, 0, stream>>>(tpc, matched, Rw, tw, done, 1);
        icp_apply_kernel<<<ptBlocks, 256, 0, stream>>>(tpc, Rw, tw, done);
        icp_err_kernel<<<1, 256, 0, stream>>>(vals, prevErr, done);
    }

    // final transform: _get_transform(p1, temp_pc), always runs
    icp_transform_kernel<<<BATCH, 256, 0, stream>>>(p1, tpc, Rw, tw, done, 0);
    icp_writeT_kernel<<<1, 64, 0, stream>>>(Rw, tw, out);
}